// Decoder_75093208203382
// MI455X (gfx1250) — compile-verified
//
#include <hip/hip_runtime.h>

typedef __attribute__((ext_vector_type(2))) float v2f;
typedef __attribute__((ext_vector_type(8))) float v8f;

#define N_D 128   // feature dim
#define N_C 4     // classes
#define BPAD 16   // B operand padded to 16 columns (N dim of WMMA tile)

// out[e, j] = relu( sum_d (r[e,d]*p[e,d]) * Wcomb[d, j] ),
// Wcomb[d, j] = sum_c w_rel[c, d] * w_cls[c, j]   (folded weights)
__global__ __launch_bounds__(256) void decoder_wmma_kernel(
    const float* __restrict__ rna,      // [N_RNA, 128]
    const float* __restrict__ prot,     // [N_PROT, 128]
    const int*   __restrict__ rna_idx,  // [E]
    const int*   __restrict__ prot_idx, // [E]
    const float* __restrict__ w_rel,    // [4, 128]
    const float* __restrict__ w_cls,    // [4, 4]
    float*       __restrict__ out,      // [E, 4]
    int E)
{
    // Combined weight, zero-padded to 16 columns so the B operand needs no
    // EXEC-divergent masking around the WMMA: w_lds[d*16 + n]
    __shared__ float w_lds[N_D * BPAD];   // 8 KB

    const int tid = threadIdx.x;
    for (int i = tid; i < N_D * BPAD; i += blockDim.x) {
        const int d = i >> 4;
        const int n = i & 15;
        float s = 0.0f;
        if (n < N_C) {
#pragma unroll
            for (int c = 0; c < N_C; ++c)
                s += w_rel[c * N_D + d] * w_cls[c * N_C + n];
        }
        w_lds[i] = s;
    }
    __syncthreads();

    const int lane        = tid & 31;
    const int wave        = tid >> 5;
    const int wave_global = blockIdx.x * (blockDim.x >> 5) + wave;
    const int e0          = wave_global * 16;
    if (e0 >= E) return;                 // wave-uniform exit: EXEC stays all-1s

    const int m     = lane & 15;         // A row (edge-in-tile) / B column
    const int kbase = (lane >> 4) << 1;  // 0 for lanes 0-15, 2 for lanes 16-31

    const int e = (e0 + m < E) ? (e0 + m) : (E - 1);   // branchless clamp

    // Each lane walks its own gathered rows; lane l and l+16 read disjoint
    // column pairs, so every feature element is fetched exactly once per wave.
    const float* rrow = rna  + (size_t)rna_idx[e]  * N_D + kbase;
    const float* prow = prot + (size_t)prot_idx[e] * N_D + kbase;

    v8f acc = {};
#pragma unroll 4
    for (int t = 0; t < N_D / 4; ++t) {
        const int d = t * 4;
        // A (16x4 f32): VGPR0 -> K=kbase, VGPR1 -> K=kbase+1
        v2f a;
        a.x = rrow[d]     * prow[d];
        a.y = rrow[d + 1] * prow[d + 1];
        // B (4x16 f32): VGPR0 -> K=kbase row, VGPR1 -> K=kbase+1 row, N=m
        v2f b;
        b.x = w_lds[(d + kbase)     * BPAD + m];
        b.y = w_lds[(d + kbase + 1) * BPAD + m];
        // D = A x B + C  (full fp32 WMMA; workload is L2-bound, fp32 is free)
        acc = __builtin_amdgcn_wmma_f32_16x16x4_f32(
                  /*neg_a=*/false, a, /*neg_b=*/false, b,
                  /*c_mod=*/(short)0, acc, /*reuse_a=*/false, /*reuse_b=*/false);
    }

    // C/D tile layout: VGPR v -> row M=v (lanes 0-15) or M=v+8 (lanes 16-31),
    // column N = lane%16. Only the first 4 columns are real classes.
    if (m < N_C) {
        const int rowoff = (lane >> 4) << 3;   // 0 or 8
        float* outp = out + (size_t)(e0 + rowoff) * N_C + m;
        if (e0 + 16 <= E) {
            // Full tile (always taken when E % 16 == 0): unguarded stores.
#pragma unroll
            for (int v = 0; v < 8; ++v) {
                const float x = acc[v];
                outp[v * N_C] = x > 0.0f ? x : 0.0f;
            }
        } else {
            // Ragged tail (defensive; unreachable for E = 500000).
#pragma unroll
            for (int v = 0; v < 8; ++v) {
                if (e0 + rowoff + v < E) {
                    const float x = acc[v];
                    outp[v * N_C] = x > 0.0f ? x : 0.0f;
                }
            }
        }
    }
}

extern "C" void kernel_launch(void* const* d_in, const int* in_sizes, int n_in,
                              void* d_out, int out_size, void* d_ws, size_t ws_size,
                              hipStream_t stream) {
    const float* rna   = (const float*)d_in[0];
    const float* prot  = (const float*)d_in[1];
    const int*   ridx  = (const int*)d_in[2];
    const int*   pidx  = (const int*)d_in[3];
    const float* wrel  = (const float*)d_in[4];
    const float* wcls  = (const float*)d_in[5];
    float*       out   = (float*)d_out;

    const int E = in_sizes[2];                 // number of edges
    const int waves_needed    = (E + 15) / 16; // one 16-edge tile per wave
    const int waves_per_block = 256 / 32;      // 8 waves / block (wave32)
    const int blocks = (waves_needed + waves_per_block - 1) / waves_per_block;

    decoder_wmma_kernel<<<blocks, 256, 0, stream>>>(
        rna, prot, ridx, pidx, wrel, wcls, out, E);
}